// MpolImage_60851096650334
// MI455X (gfx1250) — compile-verified
//
#include <hip/hip_runtime.h>
#include <math.h>

// Problem constants (match reference)
#define NPIX  4096
#define NHALF 2049          // NPIX/2 + 1
#define NVIS  500000
#define NTAPS 36

typedef float v2f __attribute__((ext_vector_type(2)));
typedef float v8f __attribute__((ext_vector_type(8)));

#define LP 65               // padded row stride for 64x64 LDS tiles (bank-conflict free)

// One K=64 real-matmul accumulation chain: 16 x V_WMMA_F32_16X16X4_F32.
// A fragment: lane<16 holds (K0,K1) of row aRow0+lane; lane>=16 holds (K2,K3).
// B fragment: VGPR0 = rows K0/K2, VGPR1 = rows K1/K3, column bCol0+(lane&15).
__device__ __forceinline__ v8f mm_chain(v8f acc, const float* __restrict__ A, int aRow0,
                                        const float* __restrict__ B, int bCol0,
                                        float bSign, int lr, int hl) {
#pragma unroll
  for (int kk = 0; kk < 16; ++kk) {
    const int k0 = kk * 4 + 2 * hl;
    v2f a, b;
    a.x = A[(aRow0 + lr) * LP + k0];
    a.y = A[(aRow0 + lr) * LP + k0 + 1];
    b.x = bSign * B[k0 * LP + bCol0 + lr];
    b.y = bSign * B[(k0 + 1) * LP + bCol0 + lr];
    acc = __builtin_amdgcn_wmma_f32_16x16x4_f32(false, a, false, b, (short)0, acc,
                                                false, false);
  }
  return acc;
}

// Precompute DFT-64 matrix (F[c][a] = exp(-2*pi*i*c*a/64), symmetric) and
// W4096 twiddle table. Runs every launch -> deterministic.
__global__ void init_tables_kernel(float* __restrict__ Fr, float* __restrict__ Fi,
                                   float2* __restrict__ Tw) {
  int i = blockIdx.x * 256 + threadIdx.x;
  if (i < 4096) {
    int c = i >> 6, a = i & 63;
    int t = (c * a) & 63;
    float ang = (float)(-6.283185307179586 * (double)t / 64.0);
    Fr[i] = cosf(ang);
    Fi[i] = sinf(ang);
    float angt = (float)(-6.283185307179586 * (double)i / 4096.0);
    Tw[i] = make_float2(cosf(angt), sinf(angt));
  }
}

// Row pass: per block, one real row of length 4096.
// x[n]=image*corrfun; Xm[a][b]=x[64a+b]; U = F*Xm (real input -> 2 chains);
// U'[c][b] = U[c][b]*W4096^{cb}; V = U'*F; X[64d+c]=V[c][d]; keep k<=2048.
__global__ __launch_bounds__(512) void fft_rows_kernel(
    const float* __restrict__ image, const float* __restrict__ corrfun,
    const float* __restrict__ gFr, const float* __restrict__ gFi,
    const float2* __restrict__ gTw, float2* __restrict__ interm) {
  __shared__ float sFr[64 * LP], sFi[64 * LP], sDr[64 * LP], sDi[64 * LP];
  const int tid = threadIdx.x;
  const int r = blockIdx.x;

  for (int i = tid; i < 4096; i += 512) {
    int c = i >> 6, a = i & 63;
    sFr[c * LP + a] = gFr[i];
    sFi[c * LP + a] = gFi[i];
  }
  const float* imr = image + (size_t)r * NPIX;
  const float* cfr = corrfun + (size_t)r * NPIX;
  for (int i = tid; i < 4096; i += 512)
    sDr[(i >> 6) * LP + (i & 63)] = imr[i] * cfr[i];
  __syncthreads();

  const int lane = tid & 31, w = tid >> 5;
  const int ti = w >> 2, tj = w & 3;          // 16 waves -> 4x4 tiles of 16x16
  const int lr = lane & 15, hl = lane >> 4;

  v8f zero = {0.f, 0.f, 0.f, 0.f, 0.f, 0.f, 0.f, 0.f};
  // Stage 1: U = F * X (X real)
  v8f ur = mm_chain(zero, sFr, ti * 16, sDr, tj * 16, 1.f, lr, hl);
  v8f ui = mm_chain(zero, sFi, ti * 16, sDr, tj * 16, 1.f, lr, hl);

  // Twiddle in registers
  float upr[8], upi[8];
#pragma unroll
  for (int v = 0; v < 8; ++v) {
    int c = ti * 16 + v + 8 * hl;
    int b = tj * 16 + lr;
    float2 t = gTw[c * b];                    // c*b <= 63*63 < 4096
    upr[v] = ur[v] * t.x - ui[v] * t.y;
    upi[v] = ur[v] * t.y + ui[v] * t.x;
  }
  __syncthreads();                            // all waves done reading sDr
#pragma unroll
  for (int v = 0; v < 8; ++v) {
    int c = ti * 16 + v + 8 * hl;
    int b = tj * 16 + lr;
    sDr[c * LP + b] = upr[v];
    sDi[c * LP + b] = upi[v];
  }
  __syncthreads();

  // Stage 2: V = U' * F ; Vr = Ur*Fr - Ui*Fi ; Vi = Ur*Fi + Ui*Fr
  v8f vr = mm_chain(zero, sDr, ti * 16, sFr, tj * 16, 1.f, lr, hl);
  vr = mm_chain(vr, sDi, ti * 16, sFi, tj * 16, -1.f, lr, hl);
  v8f vi = mm_chain(zero, sDr, ti * 16, sFi, tj * 16, 1.f, lr, hl);
  vi = mm_chain(vi, sDi, ti * 16, sFr, tj * 16, 1.f, lr, hl);

  float2* orow = interm + (size_t)r * NHALF;
#pragma unroll
  for (int v = 0; v < 8; ++v) {
    int c = ti * 16 + v + 8 * hl;
    int d = tj * 16 + lr;
    int k = 64 * d + c;
    if (k < NHALF) orow[k] = make_float2(vr[v], vi[v]);
  }
}

// Column pass: per block, one retained column k (0..2048), length-4096 complex
// FFT in place over interm[.][k], scaled by DLL^2.
__global__ __launch_bounds__(512) void fft_cols_kernel(
    const float* __restrict__ gFr, const float* __restrict__ gFi,
    const float2* __restrict__ gTw, float2* __restrict__ interm, float scale2) {
  __shared__ float sFr[64 * LP], sFi[64 * LP], sDr[64 * LP], sDi[64 * LP];
  const int tid = threadIdx.x;
  const int k = blockIdx.x;

  for (int i = tid; i < 4096; i += 512) {
    int c = i >> 6, a = i & 63;
    sFr[c * LP + a] = gFr[i];
    sFi[c * LP + a] = gFi[i];
  }
  for (int i = tid; i < 4096; i += 512) {
    float2 vv = interm[(size_t)i * NHALF + k];
    sDr[(i >> 6) * LP + (i & 63)] = vv.x;
    sDi[(i >> 6) * LP + (i & 63)] = vv.y;
  }
  __syncthreads();

  const int lane = tid & 31, w = tid >> 5;
  const int ti = w >> 2, tj = w & 3;
  const int lr = lane & 15, hl = lane >> 4;

  v8f zero = {0.f, 0.f, 0.f, 0.f, 0.f, 0.f, 0.f, 0.f};
  // Stage 1 (complex input): Ur = Fr*Xr - Fi*Xi ; Ui = Fr*Xi + Fi*Xr
  v8f ur = mm_chain(zero, sFr, ti * 16, sDr, tj * 16, 1.f, lr, hl);
  ur = mm_chain(ur, sFi, ti * 16, sDi, tj * 16, -1.f, lr, hl);
  v8f ui = mm_chain(zero, sFr, ti * 16, sDi, tj * 16, 1.f, lr, hl);
  ui = mm_chain(ui, sFi, ti * 16, sDr, tj * 16, 1.f, lr, hl);

  float upr[8], upi[8];
#pragma unroll
  for (int v = 0; v < 8; ++v) {
    int c = ti * 16 + v + 8 * hl;
    int b = tj * 16 + lr;
    float2 t = gTw[c * b];
    upr[v] = ur[v] * t.x - ui[v] * t.y;
    upi[v] = ur[v] * t.y + ui[v] * t.x;
  }
  __syncthreads();
#pragma unroll
  for (int v = 0; v < 8; ++v) {
    int c = ti * 16 + v + 8 * hl;
    int b = tj * 16 + lr;
    sDr[c * LP + b] = upr[v];
    sDi[c * LP + b] = upi[v];
  }
  __syncthreads();

  v8f vr = mm_chain(zero, sDr, ti * 16, sFr, tj * 16, 1.f, lr, hl);
  vr = mm_chain(vr, sDi, ti * 16, sFi, tj * 16, -1.f, lr, hl);
  v8f vi = mm_chain(zero, sDr, ti * 16, sFi, tj * 16, 1.f, lr, hl);
  vi = mm_chain(vi, sDi, ti * 16, sFr, tj * 16, 1.f, lr, hl);

#pragma unroll
  for (int v = 0; v < 8; ++v) {
    int c = ti * 16 + v + 8 * hl;
    int d = tj * 16 + lr;
    int k2 = 64 * d + c;                      // output frequency along axis 0
    interm[(size_t)k2 * NHALF + k] = make_float2(scale2 * vr[v], scale2 * vi[v]);
  }
}

// Sparse gather: 256 visibilities per block; C-values/indices staged through LDS
// (coalesced global reads); grid gathers are random 8B hits in L2 (67MB < 192MB).
__global__ __launch_bounds__(256) void gather_vis_kernel(
    const float* __restrict__ Cre, const float* __restrict__ Cim,
    const long long* __restrict__ Ccols, const float2* __restrict__ grid,
    float* __restrict__ out_re, float* __restrict__ out_im, int nvis) {
  __shared__ float sRe[256 * NTAPS];
  __shared__ float sIm[256 * NTAPS];
  __shared__ long long sCol[256 * NTAPS];
  const int tid = threadIdx.x;
  const int base = blockIdx.x * 256;
  const int count = min(256, nvis - base);
  const int total = count * NTAPS;
  for (int i = tid; i < total; i += 256) {
    size_t g = (size_t)base * NTAPS + i;
    sRe[i] = Cre[g];
    sIm[i] = Cim[g];
    sCol[i] = Ccols[g];
  }
  __syncthreads();
  if (tid < count) {
    float ar = 0.f, ai = 0.f;
#pragma unroll
    for (int t = 0; t < NTAPS; ++t) {
      long long c = sCol[tid * NTAPS + t];
      float2 g = grid[c];
      ar += sRe[tid * NTAPS + t] * g.x;
      ai += sIm[tid * NTAPS + t] * g.y;
    }
    out_re[base + tid] = ar;
    out_im[base + tid] = ai;
  }
}

extern "C" void kernel_launch(void* const* d_in, const int* in_sizes, int n_in,
                              void* d_out, int out_size, void* d_ws, size_t ws_size,
                              hipStream_t stream) {
  (void)in_sizes; (void)n_in; (void)out_size; (void)ws_size;
  const float* image   = (const float*)d_in[0];
  const float* corrfun = (const float*)d_in[1];
  const float* Cre     = (const float*)d_in[2];
  const float* Cim     = (const float*)d_in[3];
  const long long* Ccols = (const long long*)d_in[4];
  float* out = (float*)d_out;                  // [re(NVIS), im(NVIS)]

  // Workspace layout: Fr(16KB) | Fi(16KB) | Tw(32KB) | interm(4096*2049*8B)
  char* ws = (char*)d_ws;
  float*  Fr = (float*)ws;
  float*  Fi = Fr + 4096;
  float2* Tw = (float2*)(Fi + 4096);
  float2* interm = Tw + 4096;

  const double ARCSEC = M_PI / (180.0 * 3600.0);
  const double DLL = 0.005 * ARCSEC;
  const float scale2 = (float)(DLL * DLL);

  init_tables_kernel<<<16, 256, 0, stream>>>(Fr, Fi, Tw);
  fft_rows_kernel<<<NPIX, 512, 0, stream>>>(image, corrfun, Fr, Fi, Tw, interm);
  fft_cols_kernel<<<NHALF, 512, 0, stream>>>(Fr, Fi, Tw, interm, scale2);
  const int nblocks = (NVIS + 255) / 256;
  gather_vis_kernel<<<nblocks, 256, 0, stream>>>(Cre, Cim, Ccols, interm,
                                                 out, out + NVIS, NVIS);
}